// PManifold_27504970564314
// MI455X (gfx1250) — compile-verified
//
#include <hip/hip_runtime.h>

typedef float v2f __attribute__((ext_vector_type(2)));
typedef float v8f __attribute__((ext_vector_type(8)));

#define BB 64
#define HH 2
#define NN 512
#define KK 64
#define MM 16
#define EPSF 1e-7f
#define MAXNORM (1.0f - 1e-5f)
#define HALF_LN2 0.34657359027997264f   // 0.5 * ln(2)
#define TWO_LOG2E 2.8853900817779268f   // 2 * log2(e)

__device__ __forceinline__ float rcp_fast(float x)  { return __builtin_amdgcn_rcpf(x); }
__device__ __forceinline__ float sqrt_fast(float x) { return __builtin_amdgcn_sqrtf(x); }

// tanh for x >= 0: 1 - 2/(e^{2x}+1), via raw v_exp_f32 (exp2).
// Saturates to exactly 1.0 for large x (exp2->inf, rcp->0).
__device__ __forceinline__ float tanh_pos(float x) {
    return 1.0f - 2.0f * rcp_fast(__builtin_amdgcn_exp2f(TWO_LOG2E * x) + 1.0f);
}

// arctanh(z) = 0.5*ln((1+z)/(1-z)) = 0.5*ln2 * log2((1+z)*rcp(1-z)).
// Argument is finite/positive for z <= MAXNORM -> raw v_log_f32, no fixups.
__device__ __forceinline__ float atanh_fast(float z) {
    return HALF_LN2 * __builtin_amdgcn_logf((1.0f + z) * rcp_fast(1.0f - z));
}

// Sum over each 16-lane group via ds_swizzle group-of-32 XOR (and_mask=0x1f)
__device__ __forceinline__ float swz_sum16(float v) {
    v += __int_as_float(__builtin_amdgcn_ds_swizzle(__float_as_int(v), 0x041F)); // xor 1
    v += __int_as_float(__builtin_amdgcn_ds_swizzle(__float_as_int(v), 0x081F)); // xor 2
    v += __int_as_float(__builtin_amdgcn_ds_swizzle(__float_as_int(v), 0x101F)); // xor 4
    v += __int_as_float(__builtin_amdgcn_ds_swizzle(__float_as_int(v), 0x201F)); // xor 8
    return v;
}

// One wave (32 lanes) handles one (b, h, k') bucket.
// Block = 8 waves sharing one (b, h); grid = B*H*8 blocks.
__global__ __launch_bounds__(256)
void pmanifold_kernel(const float* __restrict__ inp,    // (B, H, N, 2)
                      const float* __restrict__ theta,  // (H, K, M)
                      float* __restrict__ out) {        // (B, H, K, M)
    __shared__ float  sT[KK * MM];   // theta[h] : 64 x 16
    __shared__ float4 sAux[KK];      // per-q: (t0, t1, ||theta_q||^2, 0)

    const int tid  = threadIdx.x;
    const int bh   = blockIdx.x >> 3;
    const int kblk = blockIdx.x & 7;
    const int b    = bh >> 1;
    const int h    = bh & 1;

    // Stage theta[h] into LDS
    for (int i = tid; i < KK * MM; i += 256)
        sT[i] = theta[h * KK * MM + i];
    __syncthreads();
    if (tid < KK) {
        float t0 = sT[tid * MM + 0];
        float t1 = sT[tid * MM + 1];
        float n2 = 0.f;
        #pragma unroll
        for (int m = 0; m < MM; ++m) { float v = sT[tid * MM + m]; n2 += v * v; }
        sAux[tid] = make_float4(t0, t1, n2, 0.f);
    }
    __syncthreads();

    const int wave = tid >> 5;
    const int lane = tid & 31;
    const int half = lane >> 4;        // which 16-lane group
    const int mcol = lane & 15;        // output component m handled by this lane
    const int kp   = kblk * 8 + wave;  // bucket k'

    // The 8 distinct diagram points feeding bucket kp: n = r*64 + kp
    const float* dg = inp + ((size_t)(b * HH + h) * NN) * 2;
    float e0[8], e1[8], s2[8], addv[8];
    #pragma unroll
    for (int r = 0; r < 8; ++r) {
        int   n  = (r << 6) + kp;
        float px = dg[2 * n], py = dg[2 * n + 1];
        float pn = fmaxf(sqrt_fast(px * px + py * py), EPSF);
        float sc = tanh_pos(pn) * rcp_fast(pn);    // exp0 at origin (2 live dims)
        e0[r] = sc * px;  e1[r] = sc * py;
        s2[r] = e0[r] * e0[r] + e1[r] * e1[r];
        addv[r] = (mcol == 0) ? e0[r] : ((mcol == 1) ? e1[r] : 0.f);
    }

    // fp32 WMMA accumulation: D = ones(16x4) x B(4x16) + C sums 4 tangent
    // rows per instruction; after 128 steps every D row equals sums[kp, :].
    v8f acc = {};
    v2f aones; aones.x = 1.0f; aones.y = 1.0f;

    #pragma unroll 4
    for (int c4 = 0; c4 < NN; c4 += 4) {
        v2f bv;
        #pragma unroll
        for (int half2 = 0; half2 < 2; ++half2) {
            // Row order inside the B fragment is irrelevant under ones-A:
            // each 4-combo group {c4..c4+3} is covered exactly once.
            int   j  = c4 + 2 * half2 + half;
            int   q  = j >> 3;          // theta row
            int   r  = j & 7;           // point index
            float4 ax = sAux[q];
            float tm  = sT[q * MM + mcol];
            // ||exp0(pad(p)) + theta_q||^2 in closed form (no cross-lane)
            float n2  = ax.z + 2.f * (e0[r] * ax.x + e1[r] * ax.y) + s2[r];
            float xn  = sqrt_fast(n2);
            float sc  = atanh_fast(fminf(xn, MAXNORM)) * rcp_fast(fmaxf(xn, EPSF));
            float tv  = sc * (tm + addv[r]);   // tangent_x[., mcol]
            if (half2 == 0) bv.x = tv; else bv.y = tv;
        }
        acc = __builtin_amdgcn_wmma_f32_16x16x4_f32(
            /*neg_a=*/false, aones, /*neg_b=*/false, bv,
            /*c_mod=*/(short)0, acc, /*reuse_a=*/false, /*reuse_b=*/false);
    }

    // acc[0] holds sums[kp, mcol] on every lane (rows replicated).
    float s   = acc[0];
    float sn2 = swz_sum16(s * s);              // 16-lane norm reduction
    float sn  = sqrt_fast(sn2);
    float snc = fmaxf(sn, EPSF);
    float th  = tanh_pos(sn);
    float ris = rcp_fast(snc);
    float xm  = th * ris * s;                  // exp0(sums)
    float xnr = th * (sn * ris);               // ||exp0(sums)||
    float y   = atanh_fast(fminf(xnr, MAXNORM)) * rcp_fast(fmaxf(xnr, EPSF)) * xm;

    if (lane < 16)
        out[(((size_t)(b * HH + h)) * KK + kp) * MM + mcol] = y;
}

extern "C" void kernel_launch(void* const* d_in, const int* in_sizes, int n_in,
                              void* d_out, int out_size, void* d_ws, size_t ws_size,
                              hipStream_t stream) {
    (void)in_sizes; (void)n_in; (void)out_size; (void)d_ws; (void)ws_size;
    const float* inp   = (const float*)d_in[0];  // (64, 2, 512, 2) f32
    const float* theta = (const float*)d_in[1];  // (2, 64, 16)     f32
    float*       out   = (float*)d_out;          // (64, 2, 64, 16) f32

    dim3 grid(BB * HH * 8);   // 1024 blocks x 8 waves = 8192 (b,h,k') waves
    dim3 block(256);
    pmanifold_kernel<<<grid, block, 0, stream>>>(inp, theta, out);
}